// GridShoot_60799557042565
// MI455X (gfx1250) — compile-verified
//
#include <hip/hip_runtime.h>
#include <math.h>

#define N     96
#define N2    (N * N)          // 9216
#define NVOX  (N * N * N)      // 884736
#define NB    2
#define ND    3
#define NC    (NB * ND)        // 6
#define NSTEPS 8

#define TWO_PI 6.28318530717958647692f

typedef __attribute__((ext_vector_type(2))) float v2f;
typedef __attribute__((ext_vector_type(8))) float v8f;
typedef __attribute__((ext_vector_type(4))) unsigned int u32x4;
typedef __attribute__((ext_vector_type(4))) int i32x4;
typedef __attribute__((ext_vector_type(8))) int i32x8;

// TDM availability: device pass only, builtin present
#if defined(__HIP_DEVICE_COMPILE__) && defined(__has_builtin)
#  if __has_builtin(__builtin_amdgcn_tensor_load_to_lds)
#    define USE_TDM 1
#  else
#    define USE_TDM 0
#  endif
#else
#  define USE_TDM 0
#endif

// ---------------------------------------------------------------------------
// helpers
// ---------------------------------------------------------------------------
__device__ __forceinline__ int wrapN(int i) { i %= N; return (i < 0) ? i + N : i; }

__device__ __forceinline__ size_t ridx(int b, int x, int y, int z, int d) {
    return ((((size_t)b * N + x) * N + y) * N + z) * ND + d;
}

// trilinear sample with periodic wrap of a (96,96,96,3) field slice f
__device__ void trilerp3(const float* __restrict__ f,
                         float px, float py, float pz, float out[3]) {
    float fx = floorf(px), fy = floorf(py), fz = floorf(pz);
    float wx = px - fx, wy = py - fy, wz = pz - fz;
    int x0 = wrapN((int)fx), y0 = wrapN((int)fy), z0 = wrapN((int)fz);
    int x1 = (x0 + 1 == N) ? 0 : x0 + 1;
    int y1 = (y0 + 1 == N) ? 0 : y0 + 1;
    int z1 = (z0 + 1 == N) ? 0 : z0 + 1;

    size_t i000 = (((size_t)x0 * N + y0) * N + z0) * ND;
    size_t i001 = (((size_t)x0 * N + y0) * N + z1) * ND;
    size_t i010 = (((size_t)x0 * N + y1) * N + z0) * ND;
    size_t i011 = (((size_t)x0 * N + y1) * N + z1) * ND;
    size_t i100 = (((size_t)x1 * N + y0) * N + z0) * ND;
    size_t i101 = (((size_t)x1 * N + y0) * N + z1) * ND;
    size_t i110 = (((size_t)x1 * N + y1) * N + z0) * ND;
    size_t i111 = (((size_t)x1 * N + y1) * N + z1) * ND;

    float w000 = (1 - wx) * (1 - wy) * (1 - wz);
    float w001 = (1 - wx) * (1 - wy) * wz;
    float w010 = (1 - wx) * wy * (1 - wz);
    float w011 = (1 - wx) * wy * wz;
    float w100 = wx * (1 - wy) * (1 - wz);
    float w101 = wx * (1 - wy) * wz;
    float w110 = wx * wy * (1 - wz);
    float w111 = wx * wy * wz;

#pragma unroll
    for (int d = 0; d < 3; d++) {
        out[d] = w000 * f[i000 + d] + w001 * f[i001 + d] +
                 w010 * f[i010 + d] + w011 * f[i011 + d] +
                 w100 * f[i100 + d] + w101 * f[i101 + d] +
                 w110 * f[i110 + d] + w111 * f[i111 + d];
    }
}

// ---------------------------------------------------------------------------
// DFT twiddle tables: cosT[k*96+n] = cos(2pi k n / 96), sinT likewise
// ---------------------------------------------------------------------------
__global__ void k_dft_tables(float* __restrict__ cosT, float* __restrict__ sinT) {
    int i = blockIdx.x * blockDim.x + threadIdx.x;
    if (i >= N2) return;
    int k = i / N, n = i % N;
    float ang = (TWO_PI / N) * (float)((k * n) % N);
    cosT[i] = cosf(ang);
    sinT[i] = sinf(ang);
}

// ---------------------------------------------------------------------------
// pack / unpack between real (B,96,96,96,3) and complex (c=b*3+d, x,y,z)
// ---------------------------------------------------------------------------
__global__ void k_pack(const float* __restrict__ f,
                       float* __restrict__ cre, float* __restrict__ cim) {
    int i = blockIdx.x * blockDim.x + threadIdx.x;
    if (i >= NVOX) return;
#pragma unroll
    for (int b = 0; b < NB; b++)
#pragma unroll
        for (int d = 0; d < ND; d++) {
            size_t ci = (size_t)(b * ND + d) * NVOX + i;
            cre[ci] = f[(size_t)b * NVOX * ND + (size_t)i * ND + d];
            cim[ci] = 0.0f;
        }
}

__global__ void k_unpack(const float* __restrict__ cre, float* __restrict__ f) {
    int i = blockIdx.x * blockDim.x + threadIdx.x;
    if (i >= NVOX) return;
#pragma unroll
    for (int b = 0; b < NB; b++)
#pragma unroll
        for (int d = 0; d < ND; d++)
            f[(size_t)b * NVOX * ND + (size_t)i * ND + d] =
                cre[(size_t)(b * ND + d) * NVOX + i];
}

// ---------------------------------------------------------------------------
// One GEMM-DFT pass along the fastest axis via V_WMMA_F32_16X16X4_F32.
// Input layout  (c, a, b, t)   index = l*96 + n  with l=(c*96+a)*96+b
// Output layout (c, t', a, b)  index = c*NVOX + k*9216 + a*96 + b
//   Yre = C*Xr + sgn*S*Xi  ;  Yim = C*Xi - sgn*S*Xr
// Block = 192 thr = 6 waves, one 16-row tile per wave, one 16-col panel per
// block.  The 16-column B panel (1536 contiguous floats per re/im) is staged
// into LDS by the Tensor Data Mover with 1-dword padding every 32 dwords
// (bank-conflict-free reads: element idx lives at LDS dword idx + (idx>>5)).
// ---------------------------------------------------------------------------
#define PANEL_DW (1536 + 1536 / 32)     // 1584 dwords incl. TDM padding

#if USE_TDM
__device__ __forceinline__ void tdm_load_1536(unsigned int lds_addr,
                                              unsigned long long ga) {
    u32x4 g0;
    g0[0] = 1u;                                   // count=1 (valid descriptor)
    g0[1] = lds_addr;                             // LDS byte address
    g0[2] = (unsigned int)(ga & 0xFFFFFFFFull);   // global_addr[31:0]
    g0[3] = (unsigned int)((ga >> 32) & 0x01FFFFFFull) | (2u << 30); // type=2
    i32x8 g1;
    g1[0] = (2 << 16) | (1 << 20) | (4 << 22);    // 4B elems, pad 1dw/32dw
    g1[1] = (1536 & 0xFFFF) << 16;                // tensor_dim0 = 1536 (lo16)
    g1[2] = (1536 >> 16) | (1 << 16);             // tensor_dim0 hi | tensor_dim1=1
    g1[3] = (1536 << 16);                         // tile_dim0 = 1536
    g1[4] = 1;                                    // tile_dim1 = 1, tile_dim2 = 0
    g1[5] = 1536;                                 // tensor_dim0_stride lo32
    g1[6] = 0;
    g1[7] = 0;
    i32x4 z4 = {0, 0, 0, 0};
#if __clang_major__ >= 23
    i32x8 z8 = {0, 0, 0, 0, 0, 0, 0, 0};
    __builtin_amdgcn_tensor_load_to_lds(g0, g1, z4, z4, z8, 0);
#else
    __builtin_amdgcn_tensor_load_to_lds(g0, g1, z4, z4, 0);
#endif
}
#endif

__global__ void __launch_bounds__(192)
k_dft_pass(const float* __restrict__ xre, const float* __restrict__ xim,
           float* __restrict__ yre, float* __restrict__ yim,
           const float* __restrict__ cosT, const float* __restrict__ sinT,
           float sgn, float scale) {
    __shared__ float pRe[PANEL_DW];
    __shared__ float pIm[PANEL_DW];

    const int lane = threadIdx.x & 31;
    const int wave = threadIdx.x >> 5;       // 0..5  -> row tile
    const int colTile = blockIdx.x;          // 0..3455
    const int m  = lane & 15;
    const int hi = lane >> 4;

    // ---- stage the 16-column B panel (re+im) into LDS --------------------
#if USE_TDM
    if (wave == 0) {
        unsigned long long gaRe =
            (unsigned long long)(uintptr_t)(xre + (size_t)colTile * 1536);
        unsigned long long gaIm =
            (unsigned long long)(uintptr_t)(xim + (size_t)colTile * 1536);
        unsigned int ldsRe = (unsigned int)(uintptr_t)(&pRe[0]);
        unsigned int ldsIm = (unsigned int)(uintptr_t)(&pIm[0]);
        tdm_load_1536(ldsRe, gaRe);
        tdm_load_1536(ldsIm, gaIm);
        __builtin_amdgcn_s_wait_tensorcnt(0);
    }
    __syncthreads();
#else
    {
        const float* gr = xre + (size_t)colTile * 1536;
        const float* gi = xim + (size_t)colTile * 1536;
        for (int t = threadIdx.x; t < 1536; t += 192) {
            int o = t + (t >> 5);            // replicate TDM pad layout
            pRe[o] = gr[t];
            pIm[o] = gi[t];
        }
        __syncthreads();
    }
#endif

    // ---- 16x16 tile GEMM: K-loop over 96 in steps of 4 -------------------
    const int rowG = wave * 16 + m;                      // A-tile row
    const float* __restrict__ ct = cosT + (size_t)rowG * N;
    const float* __restrict__ st = sinT + (size_t)rowG * N;

    v8f p1 = {}, p2 = {}, p3 = {}, p4 = {};

    for (int kk = 0; kk < N; kk += 4) {
        const int o = kk + 2 * hi;           // VGPR j <-> K = kk + j + 2*hi
        v2f ac = *(const v2f*)(ct + o);      // A tile: cos
        v2f as = *(const v2f*)(st + o);      // A tile: sin
        const int idx = m * N + o;           // panel element index
        const int l   = idx + (idx >> 5);    // padded LDS dword
        v2f br = { pRe[l], pRe[l + 1] };     // B tile: Re (pair never pads)
        v2f bi = { pIm[l], pIm[l + 1] };     // B tile: Im
        p1 = __builtin_amdgcn_wmma_f32_16x16x4_f32(false, ac, false, br, (short)0, p1, false, false);
        p2 = __builtin_amdgcn_wmma_f32_16x16x4_f32(false, as, false, bi, (short)0, p2, false, false);
        p3 = __builtin_amdgcn_wmma_f32_16x16x4_f32(false, ac, false, bi, (short)0, p3, false, false);
        p4 = __builtin_amdgcn_wmma_f32_16x16x4_f32(false, as, false, br, (short)0, p4, false, false);
    }

    // epilogue: C/D layout is M = r + 8*hi, Ncol = lane&15
    const int col = colTile * 16 + m;
    const int c  = col / N2;
    const int rm = col - c * N2;
    const int a  = rm / N;
    const int b  = rm - a * N;
    const size_t base = (size_t)c * NVOX + (size_t)a * N + b;
#pragma unroll
    for (int r = 0; r < 8; r++) {
        const int k = wave * 16 + r + 8 * hi;
        const size_t oidx = base + (size_t)k * N2;
        yre[oidx] = (p1[r] + sgn * p2[r]) * scale;
        yim[oidx] = (p3[r] - sgn * p4[r]) * scale;
    }
}

// ---------------------------------------------------------------------------
// Spectral 3x3 multiply.  A(w) = diag*I + lam*s s^T ; K = A^-1 via
// Sherman-Morrison: K = (1/diag)*I - lam/(diag*(diag+lam*|s|^2)) * s s^T.
// ---------------------------------------------------------------------------
__global__ void k_spectral(float* __restrict__ cre, float* __restrict__ cim,
                           int useInv, float extraScale) {
    int i = blockIdx.x * blockDim.x + threadIdx.x;
    if (i >= NVOX) return;
    int kx = i / N2, ky = (i / N) % N, kz = i % N;
    const float f = TWO_PI / N;
    float sx, cx, sy, cy, sz, cz;
    __sincosf(f * kx, &sx, &cx);
    __sincosf(f * ky, &sy, &cy);
    __sincosf(f * kz, &sz, &cz);

    float w    = 2.0f * (3.0f - cx - cy - cz);
    float diag = 1e-4f + 1e-3f * w + 0.2f * w * w + 0.2f * w;
    const float lam = 0.05f + 0.2f;

    float alpha, beta;
    if (useInv) {
        float ss = sx * sx + sy * sy + sz * sz;
        alpha = 1.0f / diag;
        beta  = -lam / (diag * (diag + lam * ss));
    } else {
        alpha = diag;
        beta  = lam;
    }
    alpha *= extraScale;
    beta  *= extraScale;

#pragma unroll
    for (int b = 0; b < NB; b++) {
        size_t i0 = (size_t)(b * 3 + 0) * NVOX + i;
        size_t i1 = (size_t)(b * 3 + 1) * NVOX + i;
        size_t i2 = (size_t)(b * 3 + 2) * NVOX + i;
        {
            float x0 = cre[i0], x1 = cre[i1], x2 = cre[i2];
            float dt = sx * x0 + sy * x1 + sz * x2;
            cre[i0] = alpha * x0 + beta * dt * sx;
            cre[i1] = alpha * x1 + beta * dt * sy;
            cre[i2] = alpha * x2 + beta * dt * sz;
        }
        {
            float x0 = cim[i0], x1 = cim[i1], x2 = cim[i2];
            float dt = sx * x0 + sy * x1 + sz * x2;
            cim[i0] = alpha * x0 + beta * dt * sx;
            cim[i1] = alpha * x1 + beta * dt * sy;
            cim[i2] = alpha * x2 + beta * dt * sz;
        }
    }
}

// ---------------------------------------------------------------------------
// momentum transport: m = |det J| J^T m0(psi), psi = id + idisp
// ---------------------------------------------------------------------------
__global__ void k_transport(const float* __restrict__ idisp,
                            const float* __restrict__ m0,
                            float* __restrict__ cre, float* __restrict__ cim) {
    int i = blockIdx.x * blockDim.x + threadIdx.x;
    if (i >= NVOX) return;
    int x = i / N2, y = (i / N) % N, z = i % N;
    int xp = (x + 1 == N) ? 0 : x + 1, xm = (x == 0) ? N - 1 : x - 1;
    int yp = (y + 1 == N) ? 0 : y + 1, ym = (y == 0) ? N - 1 : y - 1;
    int zp = (z + 1 == N) ? 0 : z + 1, zm = (z == 0) ? N - 1 : z - 1;

    for (int b = 0; b < NB; b++) {
        float J[3][3];
#pragma unroll
        for (int j = 0; j < 3; j++) {
            J[j][0] = 0.5f * (idisp[ridx(b, xp, y, z, j)] - idisp[ridx(b, xm, y, z, j)]);
            J[j][1] = 0.5f * (idisp[ridx(b, x, yp, z, j)] - idisp[ridx(b, x, ym, z, j)]);
            J[j][2] = 0.5f * (idisp[ridx(b, x, y, zp, j)] - idisp[ridx(b, x, y, zm, j)]);
        }
        J[0][0] += 1.0f; J[1][1] += 1.0f; J[2][2] += 1.0f;

        float px = (float)x + idisp[ridx(b, x, y, z, 0)];
        float py = (float)y + idisp[ridx(b, x, y, z, 1)];
        float pz = (float)z + idisp[ridx(b, x, y, z, 2)];
        float p[3];
        trilerp3(m0 + (size_t)b * NVOX * ND, px, py, pz, p);

        float det = J[0][0] * (J[1][1] * J[2][2] - J[1][2] * J[2][1])
                  - J[0][1] * (J[1][0] * J[2][2] - J[1][2] * J[2][0])
                  + J[0][2] * (J[1][0] * J[2][1] - J[1][1] * J[2][0]);

#pragma unroll
        for (int d = 0; d < 3; d++) {
            float mi = det * (J[0][d] * p[0] + J[1][d] * p[1] + J[2][d] * p[2]);
            size_t ci = (size_t)(b * 3 + d) * NVOX + i;
            cre[ci] = mi;
            cim[ci] = 0.0f;
        }
    }
}

// disp <- disp + v(id + disp)
__global__ void k_update_disp(float* __restrict__ disp, const float* __restrict__ v) {
    int i = blockIdx.x * blockDim.x + threadIdx.x;
    if (i >= NVOX) return;
    int x = i / N2, y = (i / N) % N, z = i % N;
    for (int b = 0; b < NB; b++) {
        float d0 = disp[ridx(b, x, y, z, 0)];
        float d1 = disp[ridx(b, x, y, z, 1)];
        float d2 = disp[ridx(b, x, y, z, 2)];
        float s[3];
        trilerp3(v + (size_t)b * NVOX * ND, (float)x + d0, (float)y + d1, (float)z + d2, s);
        disp[ridx(b, x, y, z, 0)] = d0 + s[0];
        disp[ridx(b, x, y, z, 1)] = d1 + s[1];
        disp[ridx(b, x, y, z, 2)] = d2 + s[2];
    }
}

// idispOut <- idispIn(id - v) - v
__global__ void k_update_idisp(const float* __restrict__ idin,
                               const float* __restrict__ v,
                               float* __restrict__ idout) {
    int i = blockIdx.x * blockDim.x + threadIdx.x;
    if (i >= NVOX) return;
    int x = i / N2, y = (i / N) % N, z = i % N;
    for (int b = 0; b < NB; b++) {
        float v0 = v[ridx(b, x, y, z, 0)];
        float v1 = v[ridx(b, x, y, z, 1)];
        float v2 = v[ridx(b, x, y, z, 2)];
        float s[3];
        trilerp3(idin + (size_t)b * NVOX * ND, (float)x - v0, (float)y - v1, (float)z - v2, s);
        idout[ridx(b, x, y, z, 0)] = s[0] - v0;
        idout[ridx(b, x, y, z, 1)] = s[1] - v1;
        idout[ridx(b, x, y, z, 2)] = s[2] - v2;
    }
}

// out <- grid + disp
__global__ void k_final(const float* __restrict__ disp, float* __restrict__ out) {
    int i = blockIdx.x * blockDim.x + threadIdx.x;
    if (i >= NVOX) return;
    int x = i / N2, y = (i / N) % N, z = i % N;
    for (int b = 0; b < NB; b++) {
        out[ridx(b, x, y, z, 0)] = (float)x + disp[ridx(b, x, y, z, 0)];
        out[ridx(b, x, y, z, 1)] = (float)y + disp[ridx(b, x, y, z, 1)];
        out[ridx(b, x, y, z, 2)] = (float)z + disp[ridx(b, x, y, z, 2)];
    }
}

// ---------------------------------------------------------------------------
// host orchestration
// ---------------------------------------------------------------------------
extern "C" void kernel_launch(void* const* d_in, const int* in_sizes, int n_in,
                              void* d_out, int out_size, void* d_ws, size_t ws_size,
                              hipStream_t stream) {
    const float* vel = (const float*)d_in[0];
    float* out = (float*)d_out;
    float* ws  = (float*)d_ws;

    const size_t FIELD = (size_t)NB * NVOX * ND;     // 5,308,416 floats
    const size_t CPLX  = (size_t)NC * NVOX;          // 5,308,416 floats

    size_t off = 0;
    float* cosT  = ws + off; off += N2;
    float* sinT  = ws + off; off += N2;
    float* cAre  = ws + off; off += CPLX;
    float* cAim  = ws + off; off += CPLX;
    float* cBre  = ws + off; off += CPLX;
    float* cBim  = ws + off; off += CPLX;
    float* m0    = ws + off; off += FIELD;
    float* disp  = ws + off; off += FIELD;
    float* vfld  = ws + off; off += FIELD;
    float* idA   = ws + off; off += FIELD;
    float* idB   = ws + off; off += FIELD;
    (void)ws_size; (void)in_sizes; (void)n_in; (void)out_size;

    const int EB = 256;
    const int EG = (NVOX + EB - 1) / EB;             // 3456 blocks
    const int PG = NC * N2 / 16;                     // 3456 col-tiles (blocks)

    hipMemsetAsync(disp, 0, FIELD * sizeof(float), stream);
    hipMemsetAsync(idA,  0, FIELD * sizeof(float), stream);

    k_dft_tables<<<(N2 + EB - 1) / EB, EB, 0, stream>>>(cosT, sinT);

    auto pass = [&](const float* ir, const float* ii, float* orr, float* oi,
                    float sgn, float scale) {
        k_dft_pass<<<PG, 192, 0, stream>>>(ir, ii, orr, oi, cosT, sinT, sgn, scale);
    };
    const float INVS = 1.0f / (float)N;              // 1/96 per inverse pass

    auto apply_op = [&](int useInv, float extraScale) {
        pass(cAre, cAim, cBre, cBim, +1.0f, 1.0f);
        pass(cBre, cBim, cAre, cAim, +1.0f, 1.0f);
        pass(cAre, cAim, cBre, cBim, +1.0f, 1.0f);
        k_spectral<<<EG, EB, 0, stream>>>(cBre, cBim, useInv, extraScale);
        pass(cBre, cBim, cAre, cAim, -1.0f, INVS);
        pass(cAre, cAim, cBre, cBim, -1.0f, INVS);
        pass(cBre, cBim, cAre, cAim, -1.0f, INVS);
    };

    // m0 = A v0
    k_pack<<<EG, EB, 0, stream>>>(vel, cAre, cAim);
    apply_op(0, 1.0f);
    k_unpack<<<EG, EB, 0, stream>>>(cAre, m0);

    float* idIn = idA;
    float* idOut = idB;
    for (int s = 0; s < NSTEPS; s++) {
        k_transport<<<EG, EB, 0, stream>>>(idIn, m0, cAre, cAim);
        apply_op(1, 1.0f / (float)NSTEPS);           // v = K m / STEPS
        k_unpack<<<EG, EB, 0, stream>>>(cAre, vfld);
        k_update_disp<<<EG, EB, 0, stream>>>(disp, vfld);
        k_update_idisp<<<EG, EB, 0, stream>>>(idIn, vfld, idOut);
        float* t = idIn; idIn = idOut; idOut = t;
    }

    k_final<<<EG, EB, 0, stream>>>(disp, out);
}